// Hippo_6528350289957
// MI455X (gfx1250) — compile-verified
//
#include <hip/hip_runtime.h>
#include <hip/hip_bf16.h>
#include <stdint.h>

#define T_STEPS 1024
#define BATCH   512
#define NSTATE  256
#define OUT_O   10
#define OPAD    16           // O padded to WMMA tile
#define CHUNK_K 16           // K-rows of M staged in LDS per chunk (16KB)
#define NCHUNK  (NSTATE / CHUNK_K)

typedef __attribute__((ext_vector_type(2))) float v2f;
typedef __attribute__((ext_vector_type(8))) float v8f;

// ---------------------------------------------------------------------------
// Kernel 1: M_t = (I - 0.5*dt*A)^{-1} (dense, lower-tri) and Bd_t = dt*M_t*q.
// A = -tril(q q^T, -1) - diag(n+1)  => rank-1 forward substitution, O(N)/col.
// Fully parallel over (t = blockIdx.x, column = threadIdx.x).
// ---------------------------------------------------------------------------
__global__ __launch_bounds__(NSTATE) void hippo_precompute(float* __restrict__ M,
                                                           float* __restrict__ Bd) {
  const int t = blockIdx.x;
  const int j = threadIdx.x;                    // column of M_t
  const float dt  = 1.0f / (float)(t + 1);
  const float hdt = 0.5f * dt;
  float s  = 0.0f;                              // running  sum q_k * x_k (col j)
  float sB = 0.0f;                              // running  sum q_k * y_k (Bd solve)
  float* Mt = M + (size_t)t * NSTATE * NSTATE;
  for (int n = 0; n < NSTATE; ++n) {
    float qn  = sqrtf(2.0f * (float)n + 1.0f);
    float rdn = __frcp_rn(1.0f + hdt * (float)(n + 1));
    float xn  = (((n == j) ? 1.0f : 0.0f) - hdt * qn * s) * rdn;
    s += qn * xn;
    Mt[n * NSTATE + j] = xn;                    // row-major: lane-coalesced store
    float yn = (qn - hdt * qn * sB) * rdn;      // solve L y = q (redundant per lane)
    sB += qn * yn;
    if (j == 0) Bd[t * NSTATE + n] = dt * yn;
  }
}

// ---------------------------------------------------------------------------
// Kernel 2: persistent backward scan on ONE workgroup (16 waves).
//   H_T = pad(W);  for t = T-1..0:  u_t = H*Bd_t;  H <- 2*(H x M_t) - H
// G = H x M_t done with v_wmma_f32_16x16x4_f32; M_t streamed through LDS with
// double-buffered async global->LDS loads (ASYNCcnt).
// ---------------------------------------------------------------------------
__global__ __launch_bounds__(512) void hippo_scan(const float* __restrict__ M,
                                                  const float* __restrict__ Bd,
                                                  const float* __restrict__ W,  // [10][256]
                                                  float* __restrict__ U) {      // [T][OPAD]
  __shared__ float Hs[OPAD][NSTATE];            // 16 KB
  __shared__ float Mc[2][CHUNK_K][NSTATE];      // 2 x 16 KB double buffer

  const int tid  = threadIdx.x;                 // 0..511
  const int lane = tid & 31;
  const int wave = tid >> 5;                    // 0..15, owns N-tile [16w,16w+16)
  const int m_lo = lane & 15;
  const int m_hi = lane >> 4;
  const int n0   = wave * 16;

  for (int i = tid; i < OPAD * NSTATE; i += 512) {
    int o = i / NSTATE, k = i % NSTATE;
    Hs[o][k] = (o < OUT_O) ? W[o * NSTATE + k] : 0.0f;
  }
  __syncthreads();

  #pragma unroll 1
  for (int t = T_STEPS - 1; t >= 0; --t) {
    const float* Mt = M + (size_t)t * NSTATE * NSTATE;

    // -- issue async prefetch of chunk 0 (2 x b128 per thread = 16KB total) --
    {
      unsigned lds0 = (unsigned)(uintptr_t)&Mc[0][0][0];
      #pragma unroll
      for (int r = 0; r < 2; ++r) {
        unsigned voff = (unsigned)((r * 512 + tid) * 16);
        unsigned ldso = lds0 + voff;
        asm volatile("global_load_async_to_lds_b128 %0, %1, %2"
                     :: "v"(ldso), "v"(voff), "s"(Mt) : "memory");
      }
    }

    // -- u_t = H_{t+1} . Bd_t  (wave w reduces output row w) --
    {
      const float* bd = Bd + t * NSTATE;
      float p = 0.0f;
      #pragma unroll
      for (int k = lane; k < NSTATE; k += 32) p += Hs[wave][k] * bd[k];
      #pragma unroll
      for (int off = 16; off > 0; off >>= 1) p += __shfl_down(p, off, 32);
      if (lane == 0) U[t * OPAD + wave] = p;
    }

    // -- G = H x M_t, K streamed in 16-row chunks, software pipelined --
    v8f acc = {};
    for (int c = 0; c < NCHUNK; ++c) {
      if (c + 1 < NCHUNK) {
        unsigned lds1 = (unsigned)(uintptr_t)&Mc[(c + 1) & 1][0][0];
        const float* src = Mt + (size_t)(c + 1) * CHUNK_K * NSTATE;
        #pragma unroll
        for (int r = 0; r < 2; ++r) {
          unsigned voff = (unsigned)((r * 512 + tid) * 16);
          unsigned ldso = lds1 + voff;
          asm volatile("global_load_async_to_lds_b128 %0, %1, %2"
                       :: "v"(ldso), "v"(voff), "s"(src) : "memory");
        }
        asm volatile("s_wait_asynccnt 2" ::: "memory");  // chunk c landed
      } else {
        asm volatile("s_wait_asynccnt 0" ::: "memory");
      }
      __syncthreads();

      const float (*mc)[NSTATE] = Mc[c & 1];
      const int kb = c * CHUNK_K;
      #pragma unroll
      for (int kt = 0; kt < CHUNK_K / 4; ++kt) {
        const int k0 = kt * 4;
        v2f a, b;
        // A-frag (16x4 f32): lanes 0-15 K={k0,k0+1}, lanes 16-31 K={k0+2,k0+3}
        a.x = Hs[m_lo][kb + k0 + 2 * m_hi];
        a.y = Hs[m_lo][kb + k0 + 2 * m_hi + 1];
        // B-frag (4x16 f32): row striped across lanes
        b.x = mc[k0 + m_hi][n0 + m_lo];
        b.y = mc[k0 + 2 + m_hi][n0 + m_lo];
        acc = __builtin_amdgcn_wmma_f32_16x16x4_f32(false, a, false, b,
                                                    (short)0, acc, false, false);
      }
      __syncthreads();
    }

    // -- H <- 2*G - H (each wave updates its own 16x16 N-tile) --
    float hold[8];
    #pragma unroll
    for (int r = 0; r < 8; ++r) hold[r] = Hs[r + 8 * m_hi][n0 + m_lo];
    __syncthreads();
    #pragma unroll
    for (int r = 0; r < 8; ++r)
      Hs[r + 8 * m_hi][n0 + m_lo] = 2.0f * acc[r] - hold[r];
    __syncthreads();
  }
}

// ---------------------------------------------------------------------------
// Kernel 3: out[b,o] = sum_t f[t,b] * u_t[o] + bias[o]
// ---------------------------------------------------------------------------
__global__ void hippo_head(const float* __restrict__ x,    // [T][B]
                           const float* __restrict__ U,    // [T][OPAD]
                           const float* __restrict__ bias, // [10]
                           float* __restrict__ out) {      // [B][10]
  int b = blockIdx.x * blockDim.x + threadIdx.x;
  if (b >= BATCH) return;
  float acc[OUT_O];
  #pragma unroll
  for (int o = 0; o < OUT_O; ++o) acc[o] = bias[o];
  for (int t = 0; t < T_STEPS; ++t) {
    float f = x[(size_t)t * BATCH + b];
    #pragma unroll
    for (int o = 0; o < OUT_O; ++o) acc[o] += f * U[t * OPAD + o];
  }
  #pragma unroll
  for (int o = 0; o < OUT_O; ++o) out[b * OUT_O + o] = acc[o];
}

extern "C" void kernel_launch(void* const* d_in, const int* in_sizes, int n_in,
                              void* d_out, int out_size, void* d_ws, size_t ws_size,
                              hipStream_t stream) {
  const float* inputs = (const float*)d_in[0];  // [1024,512,1]
  const float* mlp_w  = (const float*)d_in[1];  // [10,256]
  const float* mlp_b  = (const float*)d_in[2];  // [10]
  float* out = (float*)d_out;                   // [512,10]

  // Workspace layout (needs ~270 MB): M [T][N][N], Bd [T][N], U [T][OPAD]
  float* d_M  = (float*)d_ws;
  float* d_Bd = d_M  + (size_t)T_STEPS * NSTATE * NSTATE;
  float* d_U  = d_Bd + (size_t)T_STEPS * NSTATE;

  hippo_precompute<<<T_STEPS, NSTATE, 0, stream>>>(d_M, d_Bd);
  hippo_scan<<<1, 512, 0, stream>>>(d_M, d_Bd, mlp_w, d_U);
  hippo_head<<<(BATCH + 255) / 256, 256, 0, stream>>>(inputs, d_U, mlp_b, out);
}